// GATBlock_57286273794488
// MI455X (gfx1250) — compile-verified
//
#include <hip/hip_runtime.h>
#include <math.h>

#define NNODES   50000
#define NEDGES   800000
#define KDIM     128     // both layers: input dim == 128
#define FDIM     128     // H*OUT
#define HD       2
#define OUTD     64
#define NEG_SLOPE 0.2f

typedef __attribute__((ext_vector_type(2))) float v2f;
typedef __attribute__((ext_vector_type(8))) float v8f;

#ifndef __has_builtin
#define __has_builtin(x) 0
#endif
#if __has_builtin(__builtin_amdgcn_wmma_f32_16x16x4_f32)
#define HAVE_WMMA_F32 1
#else
#define HAVE_WMMA_F32 0
#endif

// ---------------------------------------------------------------------------
// GEMM: F[N,128] = A[N,128] x W[128,128], full f32 via V_WMMA_F32_16X16X4_F32.
// One wave per 16-row stripe; 8 accumulators cover all 128 output columns.
// f32 WMMA layouts (ISA 7.12.2):
//   A 16x4 : lane half h, lane l -> M=l, VGPR v holds K = 2h+v
//   B 4x16 : lane half h, lane l -> N=l, VGPR v holds K = 2h+v
//   C 16x16: VGPR v, lane half h, lane l -> M = v + 8h, N = l
// ---------------------------------------------------------------------------
__global__ __launch_bounds__(32) void k_gemm(const float* __restrict__ A,
                                             const float* __restrict__ W,
                                             float* __restrict__ F) {
#if HAVE_WMMA_F32
  const int lane = threadIdx.x;
  const int half = lane >> 4;
  const int l    = lane & 15;
  const int m0   = blockIdx.x * 16;
  v8f c[8] = {};
  const float* __restrict__ arow = A + (size_t)(m0 + l) * KDIM;
  for (int kb = 0; kb < KDIM; kb += 4) {
    const int ka = kb + 2 * half;
    v2f a;
    a.x = arow[ka];
    a.y = arow[ka + 1];
#pragma unroll
    for (int t = 0; t < 8; ++t) {
      v2f b;
      b.x = W[(size_t)ka * FDIM + t * 16 + l];
      b.y = W[(size_t)(ka + 1) * FDIM + t * 16 + l];
      c[t] = __builtin_amdgcn_wmma_f32_16x16x4_f32(
          /*neg_a=*/false, a, /*neg_b=*/false, b,
          /*c_mod=*/(short)0, c[t], /*reuse_a=*/false, /*reuse_b=*/false);
    }
  }
#pragma unroll
  for (int t = 0; t < 8; ++t)
#pragma unroll
    for (int v = 0; v < 8; ++v)
      F[(size_t)(m0 + v + 8 * half) * FDIM + t * 16 + l] = c[t][v];
#else
  // Fallback VALU path (compile-safety only).
  const int lane = threadIdx.x;
  const int m0   = blockIdx.x * 16;
  for (int rr = 0; rr < 16; ++rr) {
    const float* arow = A + (size_t)(m0 + rr) * KDIM;
    for (int c0 = lane; c0 < FDIM; c0 += 32) {
      float acc = 0.f;
      for (int k = 0; k < KDIM; ++k) acc += arow[k] * W[(size_t)k * FDIM + c0];
      F[(size_t)(m0 + rr) * FDIM + c0] = acc;
    }
  }
#endif
}

// ---------------------------------------------------------------------------
// Attention scores: el/er[n,h] = sum_o f[n,h,o] * al/ar[h,o]
// ---------------------------------------------------------------------------
__global__ void k_scores(const float* __restrict__ F,
                         const float* __restrict__ al,
                         const float* __restrict__ ar,
                         float* __restrict__ el, float* __restrict__ er) {
  const int idx = blockIdx.x * blockDim.x + threadIdx.x;
  if (idx >= NNODES * HD) return;
  const int n = idx >> 1, h = idx & 1;
  const float* frow = F + (size_t)n * FDIM + h * OUTD;
  const float* alr  = al + h * OUTD;
  const float* arr  = ar + h * OUTD;
  float sl = 0.f, sr = 0.f;
#pragma unroll 8
  for (int o = 0; o < OUTD; ++o) {
    const float v = frow[o];
    sl += v * alr[o];
    sr += v * arr[o];
  }
  el[idx] = sl;
  er[idx] = sr;
}

// Order-preserving float<->uint for atomic max (0u == below all encodings).
__device__ __forceinline__ unsigned f2ord(float x) {
  unsigned u = __float_as_uint(x);
  return ((int)u >= 0) ? (u | 0x80000000u) : ~u;
}
__device__ __forceinline__ float ord2f(unsigned u) {
  return (u & 0x80000000u) ? __uint_as_float(u & 0x7FFFFFFFu) : __uint_as_float(~u);
}

// Pass 1: e = leaky_relu(el[src]+er[dst]); segment-max into mord[dst].
__global__ void k_edge_max(const int* __restrict__ src, const int* __restrict__ dst,
                           const float* __restrict__ el, const float* __restrict__ er,
                           float* __restrict__ esc, unsigned* __restrict__ mord) {
  const int e = blockIdx.x * blockDim.x + threadIdx.x;
  if (e >= NEDGES) return;
  const int s = src[e], d = dst[e];
#pragma unroll
  for (int h = 0; h < HD; ++h) {
    float v = el[s * HD + h] + er[d * HD + h];
    v = (v >= 0.f) ? v : NEG_SLOPE * v;
    esc[(size_t)e * HD + h] = v;
    atomicMax(&mord[d * HD + h], f2ord(v));
  }
}

// Pass 2: ex = exp(e - m[dst]); segment-sum into ssum[dst].
__global__ void k_edge_exp(const int* __restrict__ dst,
                           const unsigned* __restrict__ mord,
                           float* __restrict__ esc, float* __restrict__ ssum) {
  const int e = blockIdx.x * blockDim.x + threadIdx.x;
  if (e >= NEDGES) return;
  const int d = dst[e];
#pragma unroll
  for (int h = 0; h < HD; ++h) {
    const float m  = ord2f(mord[d * HD + h]);
    const float ex = expf(esc[(size_t)e * HD + h] - m);
    esc[(size_t)e * HD + h] = ex;
    atomicAdd(&ssum[d * HD + h], ex);
  }
}

// Pass 3: one wave per edge. Lane j gathers float4 of f[src], scales by the
// edge's softmax weight for its head (head = j>>4), scatter-adds into agg[dst].
__global__ __launch_bounds__(256) void k_edge_msg(const int* __restrict__ src,
                                                  const int* __restrict__ dst,
                                                  const float* __restrict__ esc,
                                                  const float* __restrict__ ssum,
                                                  const float* __restrict__ F,
                                                  float* __restrict__ agg) {
  const int wave = blockIdx.x * 8 + (threadIdx.x >> 5);
  const int lane = threadIdx.x & 31;
  if (wave >= NEDGES) return;
  const int s = src[wave], d = dst[wave];
  const int h = lane >> 4;  // lane*4 in [0,64) -> head 0, [64,128) -> head 1
  const float a = esc[(size_t)wave * HD + h] / ssum[d * HD + h];
  const float4 v = *(const float4*)(F + (size_t)s * FDIM + lane * 4);
  float* out = agg + (size_t)d * FDIM + lane * 4;
  atomicAdd(out + 0, v.x * a);
  atomicAdd(out + 1, v.y * a);
  atomicAdd(out + 2, v.z * a);
  atomicAdd(out + 3, v.w * a);
}

// acc += elu(agg + b)
__global__ void k_accum(const float* __restrict__ agg, const float* __restrict__ b,
                        float* __restrict__ acc) {
  const int i = blockIdx.x * blockDim.x + threadIdx.x;
  if (i >= NNODES * FDIM) return;
  const float x = agg[i] + b[i & (FDIM - 1)];
  acc[i] += (x > 0.f) ? x : (expf(x) - 1.f);
}

// ---------------------------------------------------------------------------
static void run_layer(const float* hin, const float* Wm, const float* al,
                      const float* ar, const float* bb, const int* srcR,
                      const int* dstR, float* acc, float* f, float* agg,
                      float* el, float* er, unsigned* mord, float* ssum,
                      float* esc, hipStream_t stream) {
  hipMemsetAsync(acc, 0, (size_t)NNODES * FDIM * sizeof(float), stream);
  for (int r = 0; r < 2; ++r) {
    k_gemm<<<NNODES / 16, 32, 0, stream>>>(hin, Wm + (size_t)r * KDIM * FDIM, f);
    k_scores<<<(NNODES * HD + 255) / 256, 256, 0, stream>>>(
        f, al + r * HD * OUTD, ar + r * HD * OUTD, el, er);
    hipMemsetAsync(mord, 0, (size_t)NNODES * HD * sizeof(unsigned), stream);
    hipMemsetAsync(ssum, 0, (size_t)NNODES * HD * sizeof(float), stream);
    hipMemsetAsync(agg, 0, (size_t)NNODES * FDIM * sizeof(float), stream);
    const int* sr = srcR + (size_t)r * NEDGES;
    const int* dr = dstR + (size_t)r * NEDGES;
    k_edge_max<<<(NEDGES + 255) / 256, 256, 0, stream>>>(sr, dr, el, er, esc, mord);
    k_edge_exp<<<(NEDGES + 255) / 256, 256, 0, stream>>>(dr, mord, esc, ssum);
    k_edge_msg<<<(NEDGES + 7) / 8, 256, 0, stream>>>(sr, dr, esc, ssum, f, agg);
    k_accum<<<(NNODES * FDIM + 255) / 256, 256, 0, stream>>>(agg, bb + r * FDIM, acc);
  }
}

extern "C" void kernel_launch(void* const* d_in, const int* in_sizes, int n_in,
                              void* d_out, int out_size, void* d_ws, size_t ws_size,
                              hipStream_t stream) {
  const float* x   = (const float*)d_in[0];
  const float* W0  = (const float*)d_in[1];
  const float* al0 = (const float*)d_in[2];
  const float* ar0 = (const float*)d_in[3];
  const float* b0  = (const float*)d_in[4];
  const float* W1  = (const float*)d_in[5];
  const float* al1 = (const float*)d_in[6];
  const float* ar1 = (const float*)d_in[7];
  const float* b1  = (const float*)d_in[8];
  const int* src0  = (const int*)d_in[9];
  const int* dst0  = (const int*)d_in[10];
  const int* src1  = (const int*)d_in[11];
  const int* dst1  = (const int*)d_in[12];

  float* ws = (float*)d_ws;
  float*    f    = ws;            ws += (size_t)NNODES * FDIM;
  float*    agg  = ws;            ws += (size_t)NNODES * FDIM;
  float*    h1   = ws;            ws += (size_t)NNODES * FDIM;
  float*    el   = ws;            ws += (size_t)NNODES * HD;
  float*    er   = ws;            ws += (size_t)NNODES * HD;
  unsigned* mord = (unsigned*)ws; ws += (size_t)NNODES * HD;
  float*    ssum = ws;            ws += (size_t)NNODES * HD;
  float*    esc  = ws;            // NEDGES * HD floats

  // Layer 1: x -> h1
  run_layer(x, W0, al0, ar0, b0, src0, dst0, h1, f, agg, el, er, mord, ssum, esc, stream);
  // Layer 2: h1 -> d_out
  run_layer(h1, W1, al1, ar1, b1, src1, dst1, (float*)d_out, f, agg, el, er, mord, ssum, esc, stream);
}